// AttCnn_48610439856236
// MI455X (gfx1250) — compile-verified
//
#include <hip/hip_runtime.h>

typedef __attribute__((ext_vector_type(2))) float v2f;
typedef __attribute__((ext_vector_type(8))) float v8f;

#define BB   16
#define NF   64
#define EC   3
#define HH   256
#define WW   256
#define HW   (HH * WW)        // 65536
#define NF1  8
#define NF2  16
#define NPIX (BB * HW)        // 1048576
#define NTILES (NPIX / 16)    // 65536

// LDS layout (float offsets)
#define OFF_W1 0
#define OFF_B1 24
#define OFF_W2 32
#define OFF_B2 160
#define OFF_WM 176
#define OFF_BM 1200
#define OFF_WA 1264
#define OFF_BA 2288
#define SMEM_FLOATS 2352

__device__ __forceinline__ float leaky(float x) { return x >= 0.0f ? x : 0.2f * x; }

__global__ __launch_bounds__(256) void attcnn_wmma_kernel(
    const float* __restrict__ fm, const float* __restrict__ ex,
    const float* __restrict__ w1, const float* __restrict__ b1,
    const float* __restrict__ w2, const float* __restrict__ b2,
    const float* __restrict__ wm, const float* __restrict__ bm,
    const float* __restrict__ wa, const float* __restrict__ ba,
    float* __restrict__ out)
{
    __shared__ float sm[SMEM_FLOATS];
    const int tid = threadIdx.x;
    for (int i = tid; i < 24;   i += 256) sm[OFF_W1 + i] = w1[i];
    for (int i = tid; i < 8;    i += 256) sm[OFF_B1 + i] = b1[i];
    for (int i = tid; i < 128;  i += 256) sm[OFF_W2 + i] = w2[i];
    for (int i = tid; i < 16;   i += 256) sm[OFF_B2 + i] = b2[i];
    for (int i = tid; i < 1024; i += 256) sm[OFF_WM + i] = wm[i];
    for (int i = tid; i < 64;   i += 256) sm[OFF_BM + i] = bm[i];
    for (int i = tid; i < 1024; i += 256) sm[OFF_WA + i] = wa[i];
    for (int i = tid; i < 64;   i += 256) sm[OFF_BA + i] = ba[i];
    __syncthreads();

    const int lane = tid & 31;
    const int hi   = lane >> 4;   // lane half: selects K/M sub-rows per WMMA layout
    const int li   = lane & 15;   // pixel within tile (lanes 16..31 mirror 0..15)

    // ---- hoisted loop-invariant fragments (tiny: 3 x v2f + 8 floats) ----
    // Stage-1 A: rows j<8 = [w1[j,0..2] | b1[j]] (bias folded in as K=3), rows 8..15 = 0
    v2f a1;
    if (li < NF1) {
        a1.x = hi ? sm[OFF_W1 + 3 * li + 2] : sm[OFF_W1 + 3 * li + 0];
        a1.y = hi ? sm[OFF_B1 + li]         : sm[OFF_W1 + 3 * li + 1];
    } else {
        a1.x = 0.0f; a1.y = 0.0f;
    }
    // Stage-2 A: w2 (16x8) as two 16x4 K-tiles
    v2f a2[2];
    #pragma unroll
    for (int kt = 0; kt < 2; ++kt) {
        a2[kt].x = sm[OFF_W2 + li * NF1 + 4 * kt + 2 * hi + 0];
        a2[kt].y = sm[OFF_W2 + li * NF1 + 4 * kt + 2 * hi + 1];
    }
    // Stage-2 bias in D layout (VGPR r <-> row r + 8*hi)
    float bias2[8];
    #pragma unroll
    for (int r = 0; r < 8; ++r) bias2[r] = sm[OFF_B2 + r + 8 * hi];

    const int wave   = (blockIdx.x * 256 + tid) >> 5;
    const int nwaves = (gridDim.x * 256) >> 5;

    for (int tile = wave; tile < NTILES; tile += nwaves) {
        const int p0 = tile << 4;
        const int b  = p0 >> 16;            // HW == 65536
        const int hw = (p0 & (HW - 1)) + li;

        // prefetch next tile's streams (global_prefetch_b8)
        const int tn = tile + nwaves;
        if (tn < NTILES) {
            const int p0n = tn << 4;
            const int bn  = p0n >> 16;
            const int hwn = (p0n & (HW - 1)) + li;
            __builtin_prefetch(ex + bn * EC * HW + hwn, 0, 1);
            __builtin_prefetch(fm + bn * NF * HW + hwn, 0, 1);
        }

        // ---- stage 1: fea1 = leaky(w1 @ e + b1) as one WMMA ----
        const float* ep = ex + b * EC * HW + hw;
        const float e0 = ep[0], e1 = ep[HW], e2 = ep[2 * HW];
        // B = [e0; e1; e2; 1] (4x16): VGPR0 = K{0|2}, VGPR1 = K{1|3}
        v2f be;
        be.x = hi ? e2   : e0;
        be.y = hi ? 1.0f : e1;
        v8f d1 = {0, 0, 0, 0, 0, 0, 0, 0};
        d1 = __builtin_amdgcn_wmma_f32_16x16x4_f32(
            false, a1, false, be, (short)0, d1, false, false);
        #pragma unroll
        for (int r = 0; r < 8; ++r) d1[r] = leaky(d1[r]);

        // ---- stage 2: fea2 = leaky(w2 @ fea1 + b2), two WMMAs ----
        // fea1 rows 0..7 live in VGPRs 0..7 of lanes 0..15; rebuild B via shfl
        v8f d2;
        #pragma unroll
        for (int r = 0; r < 8; ++r) d2[r] = bias2[r];
        #pragma unroll
        for (int kt = 0; kt < 2; ++kt) {
            const float r0 = __shfl(d1[4 * kt + 0], li, 32);
            const float r1 = __shfl(d1[4 * kt + 1], li, 32);
            const float r2 = __shfl(d1[4 * kt + 2], li, 32);
            const float r3 = __shfl(d1[4 * kt + 3], li, 32);
            v2f b2f;
            b2f.x = hi ? r2 : r0;
            b2f.y = hi ? r3 : r1;
            d2 = __builtin_amdgcn_wmma_f32_16x16x4_f32(
                false, a2[kt], false, b2f, (short)0, d2, false, false);
        }
        #pragma unroll
        for (int r = 0; r < 8; ++r) d2[r] = leaky(d2[r]);

        // ---- rebuild fea2 as four 4x16 B fragments ----
        // rows 0..7 -> VGPR r, lanes 0..15 ; rows 8..15 -> VGPR r, lanes 16..31
        v2f bf[4];
        #pragma unroll
        for (int kt = 0; kt < 4; ++kt) {
            const int src = (kt < 2) ? li : (li + 16);
            const int rb  = (kt & 1) * 4;
            const float r0 = __shfl(d2[rb + 0], src, 32);
            const float r1 = __shfl(d2[rb + 1], src, 32);
            const float r2 = __shfl(d2[rb + 2], src, 32);
            const float r3 = __shfl(d2[rb + 3], src, 32);
            bf[kt].x = hi ? r2 : r0;
            bf[kt].y = hi ? r3 : r1;
        }

        const float* fmb = fm  + b * NF * HW + hw;
        float*       ob  = out + b * NF * HW + hw;

        // ---- mul/add matvecs: 4 channel tiles x 4 K-tiles x 2 matrices ----
        #pragma unroll 1
        for (int m = 0; m < 4; ++m) {
            v8f amul, aadd;
            #pragma unroll
            for (int r = 0; r < 8; ++r) {
                amul[r] = sm[OFF_BM + 16 * m + r + 8 * hi];
                aadd[r] = sm[OFF_BA + 16 * m + r + 8 * hi];
            }
            #pragma unroll
            for (int kt = 0; kt < 4; ++kt) {
                const int aidx = ((16 * m + li) << 4) + 4 * kt + 2 * hi;
                const v2f awm = *(const v2f*)&sm[OFF_WM + aidx];
                const v2f awa = *(const v2f*)&sm[OFF_WA + aidx];
                amul = __builtin_amdgcn_wmma_f32_16x16x4_f32(
                    false, awm, false, bf[kt], (short)0, amul, false, false);
                aadd = __builtin_amdgcn_wmma_f32_16x16x4_f32(
                    false, awa, false, bf[kt], (short)0, aadd, false, false);
            }
            #pragma unroll
            for (int r = 0; r < 8; ++r) {
                const int c  = 16 * m + r + 8 * hi;
                const float fv = fmb[c * HW];
                const float o  = aadd[r] * __builtin_amdgcn_rcpf(1.0f + fv * amul[r]);
                ob[c * HW] = o;
            }
        }
    }
}

extern "C" void kernel_launch(void* const* d_in, const int* in_sizes, int n_in,
                              void* d_out, int out_size, void* d_ws, size_t ws_size,
                              hipStream_t stream) {
    attcnn_wmma_kernel<<<1024, 256, 0, stream>>>(
        (const float*)d_in[0],  // feature_maps
        (const float*)d_in[1],  // extra_maps
        (const float*)d_in[2],  // w1
        (const float*)d_in[3],  // b1
        (const float*)d_in[4],  // w2
        (const float*)d_in[5],  // b2
        (const float*)d_in[6],  // wm
        (const float*)d_in[7],  // bm
        (const float*)d_in[8],  // wa
        (const float*)d_in[9],  // ba
        (float*)d_out);
}